// CCNN1d_55018531061965
// MI455X (gfx1250) — compile-verified
//
#include <hip/hip_runtime.h>
#include <cstdint>

// Problem shape (fixed by reference)
constexpr int BATCH = 32;
constexpr int TSTEPS = 64;
constexpr int L = 8192;

// Tiling: chunk C computed per workgroup, halo H each side (H >= T covers the
// 1-element/step dependency cone exactly), window W = C + 2H.
constexpr int C = 512;
constexpr int H = 64;
constexpr int W = C + 2 * H;      // 640 window elements
constexpr int NT = 320;           // 10 wave32 per workgroup
constexpr int ELEMS = W / NT;     // 2 window elements per thread
constexpr int SEGS = W / 4;       // 160 x b128 async segments per step

__global__ __launch_bounds__(NT) void pcnn_step_kernel(
    const float* __restrict__ x,  // [B, T, L]
    const float* __restrict__ w,  // [3]
    float* __restrict__ out)      // [B, T, L]
{
    // Ping-pong y buffers, each with zero guard cells at [0] and [W+1]
    // (conv zero padding at true array edges; halo edges discarded by design).
    __shared__ float ybuf[2][W + 2];
    __shared__ float xbuf[2][W];  // double-buffered x window (async DMA target)

    const int tid    = threadIdx.x;
    const int chunk  = blockIdx.x;
    const int b      = blockIdx.y;
    const int cstart = chunk * C;

    const float df = 0.90483741803596f;  // exp(-ALPHA_F), ALPHA_F=0.1
    const float dl = 0.36787944117144f;  // exp(-ALPHA_L), ALPHA_L=1.0
    const float de = 0.36787944117144f;  // exp(-ALPHA_E), ALPHA_E=1.0
    const float w0 = w[0], w1 = w[1], w2 = w[2];

    // zero both y buffers (guards included)
    for (int i = tid; i < 2 * (W + 2); i += NT) ybuf[0][i] = 0.0f;

    float f[ELEMS], lst[ELEMS], e[ELEMS], y[ELEMS];
    bool  valid[ELEMS];
#pragma unroll
    for (int k = 0; k < ELEMS; ++k) {
        const int wpos = tid + k * NT;
        const int g    = cstart - H + wpos;
        valid[k] = (g >= 0) && (g < L);
        f[k] = 0.0f; lst[k] = 0.0f; e[k] = 10.0f; y[k] = 0.0f;  // e0 = V_E/ALPHA_E
    }

    // ---- async x prefetch setup (lanes 0..SEGS-1 each move 16B per step) ----
    const uint32_t lds_x0 = (uint32_t)(uintptr_t)(&xbuf[0][0]) + (uint32_t)tid * 16u;
    const uint32_t lds_x1 = (uint32_t)(uintptr_t)(&xbuf[1][0]) + (uint32_t)tid * 16u;
    // Segment's element index within the row; clamp so boundary chunks never
    // read out of the allocation (clamped segments land only on halo slots
    // whose y is forced to 0, so garbage x there is harmless).
    int gseg = cstart - H + tid * 4;
    if (gseg < 0) gseg = 0;
    if (gseg > L - 4) gseg = L - 4;
    const uint32_t voff = (uint32_t)(gseg * 4);  // byte offset within a row
    const uint64_t xrow0 = (uint64_t)(uintptr_t)(x + (size_t)b * TSTEPS * L);

    // Prime the pipeline: t=0 window -> xbuf[0]
    if (tid < SEGS) {
        asm volatile("global_load_async_to_lds_b128 %0, %1, %2"
                     :: "v"(lds_x0), "v"(voff), "s"(xrow0) : "memory");
    }

    for (int t = 0; t < TSTEPS; ++t) {
        // Prefetch next step's window into the other buffer, then retire the
        // current one. Async loads complete in order, so asynccnt<=1 means the
        // step-t transfer has fully landed in LDS for this wave.
        if (t + 1 < TSTEPS) {
            if (tid < SEGS) {
                const uint64_t basen = xrow0 + (uint64_t)(t + 1) * (uint64_t)(L * 4);
                const uint32_t ldsn  = ((t + 1) & 1) ? lds_x1 : lds_x0;
                asm volatile("global_load_async_to_lds_b128 %0, %1, %2"
                             :: "v"(ldsn), "v"(voff), "s"(basen) : "memory");
            }
            asm volatile("s_wait_asynccnt 1" ::: "memory");
        } else {
            asm volatile("s_wait_asynccnt 0" ::: "memory");
        }
        // Single barrier per step:
        //  - xbuf[t&1] landed for every wave (each wave waited its own asynccnt)
        //  - step t-1's writes to ybuf[t&1] are visible
        //  - step t-1's reads of ybuf[(t+1)&1] are complete, so we may overwrite
        __syncthreads();

        const float* xc = xbuf[t & 1];
        const float* yprev = ybuf[t & 1];        // y_{t-1}, guarded
        float*       ynext = ybuf[(t + 1) & 1];  // y_t destination
#pragma unroll
        for (int k = 0; k < ELEMS; ++k) {
            const int wpos = tid + k * NT;
            const float yl = yprev[wpos];      // y_{t-1}[wpos-1]
            const float yr = yprev[wpos + 2];  // y_{t-1}[wpos+1]
            const float yc = y[k];             // y_{t-1}[wpos] from register
            const float xv = xc[wpos];
            f[k]   = df * f[k] + xv + (w0 * yl + w1 * yc + w2 * yr);
            lst[k] = dl * lst[k] + yl + yr;                 // fixed [1,0,1] kernel
            const float u = f[k] * (1.0f + 0.5f * lst[k]);
            e[k]   = de * e[k] + 10.0f * yc;                // V_E * y_prev
            const float yn = 1.0f / (1.0f + __expf(e[k] - u));  // sigmoid(u-e)
            y[k] = valid[k] ? yn : 0.0f;       // zero padding beyond [0, L)
            ynext[wpos + 1] = y[k];
            if (wpos >= H && wpos < H + C) {   // interior of the window only
                const int g = cstart + (wpos - H);
                out[((size_t)b * TSTEPS + t) * L + g] = y[k];
            }
        }
    }
}

extern "C" void kernel_launch(void* const* d_in, const int* in_sizes, int n_in,
                              void* d_out, int out_size, void* d_ws, size_t ws_size,
                              hipStream_t stream) {
    (void)in_sizes; (void)n_in; (void)out_size; (void)d_ws; (void)ws_size;
    const float* x = (const float*)d_in[0];
    const float* w = (const float*)d_in[1];
    float* out = (float*)d_out;
    dim3 grid(L / C, BATCH);   // 16 chunks x 32 batches = 512 workgroups
    pcnn_step_kernel<<<grid, NT, 0, stream>>>(x, w, out);
}